// Attention_5626407157951
// MI455X (gfx1250) — compile-verified
//
#include <hip/hip_runtime.h>
#include <math.h>

// Problem constants (match reference)
#define H      32
#define D      128
#define HID    4096
#define BS     64
#define B_SZ   32
#define NBLK   16
#define MAX_KV (NBLK * BS)
#define NCOL   (3 * HID)        // 12288
#define ROPE_BASE 10000.0f

typedef float v2f __attribute__((ext_vector_type(2)));
typedef float v8f __attribute__((ext_vector_type(8)));

// ---------------------------------------------------------------------------
// Kernel 1: qkv = X(32x4096) @ W_pack(4096x12288), fp32 WMMA 16x16x4.
// One wave per 16x32 output strip (one m-tile x two adjacent n-tiles):
//  - B reads cover a full 128B cacheline of each W row per wave
//  - A-frag reused across the two WMMAs (2 wmma per A load)
// 2 m-tiles x 384 n-pairs = 768 waves -> 96 blocks x 8 waves.
// A-frag (16x4): lanes 0-15 hold M=0..15 K={0,1}; lanes 16-31 K={2,3}.
// B-frag (4x16): lanes 0-15 hold N=0..15 K={0,1}; lanes 16-31 K={2,3}.
// C/D (16x16):  VGPR r: lanes 0-15 -> M=r, lanes 16-31 -> M=r+8.
// ---------------------------------------------------------------------------
__global__ void qkv_gemm_wmma(const float* __restrict__ X,
                              const float* __restrict__ W,
                              float* __restrict__ Y)
{
    const int wave  = (blockIdx.x * blockDim.x + threadIdx.x) >> 5;
    const int lane  = threadIdx.x & 31;
    const int mtile = wave & 1;         // 0..1
    const int npair = wave >> 1;        // 0..383 -> columns [32*npair, 32*npair+31]
    const int mn    = lane & 15;        // row for A, col-within-tile for B
    const int kk    = (lane >> 4) << 1; // 0 or 2

    const float* __restrict__ arow  = X + (size_t)(mtile * 16 + mn) * HID;
    const float* __restrict__ bcol0 = W + (size_t)npair * 32 + mn;       // n-tile 0
    const float* __restrict__ bcol1 = bcol0 + 16;                        // n-tile 1

    v8f c0 = {};
    v8f c1 = {};
    for (int k0 = 0; k0 < HID; k0 += 4) {
        // prefetch the weight stream ~64 k ahead (128B line covers both tiles)
        int kpf = k0 + 64 < HID ? k0 + 64 : HID - 4;
        __builtin_prefetch(bcol0 + (size_t)kpf * NCOL, 0, 1);

        v2f a = *(const v2f*)(arow + k0 + kk);
        v2f b0, b1;
        b0.x = bcol0[(size_t)(k0 + kk)     * NCOL];
        b0.y = bcol0[(size_t)(k0 + kk + 1) * NCOL];
        b1.x = bcol1[(size_t)(k0 + kk)     * NCOL];
        b1.y = bcol1[(size_t)(k0 + kk + 1) * NCOL];
        c0 = __builtin_amdgcn_wmma_f32_16x16x4_f32(false, a, false, b0,
                                                   (short)0, c0, false, false);
        c1 = __builtin_amdgcn_wmma_f32_16x16x4_f32(false, a, false, b1,
                                                   (short)0, c1, false, false);
    }

    const int rbase = mtile * 16 + ((lane >> 4) << 3);
    float* __restrict__ yp = Y + (size_t)npair * 32 + mn;
#pragma unroll
    for (int r = 0; r < 8; ++r) {
        yp[(size_t)(rbase + r) * NCOL]      = c0[r];
        yp[(size_t)(rbase + r) * NCOL + 16] = c1[r];
    }
}

// ---------------------------------------------------------------------------
// Kernel 2: fused RoPE + paged attention, one wave (32 threads) per (b, h).
// q/k get RoPE in LDS; the fresh k/v row replaces the cache read at s == pos
// (functional cache update -> inputs are never mutated).
// ---------------------------------------------------------------------------
__global__ void attn_kernel(const float* __restrict__ qkv,
                            float* __restrict__ attn_out,
                            const float* __restrict__ k_cache,
                            const float* __restrict__ v_cache,
                            const int* __restrict__ hist,
                            const int* __restrict__ boff)
{
    __shared__ __align__(16) float qs[D];
    __shared__ __align__(16) float ks[D];
    __shared__ __align__(16) float vs[D];
    __shared__ float probs[MAX_KV];

    const int b    = blockIdx.x / H;
    const int h    = blockIdx.x % H;
    const int lane = threadIdx.x;
    const int pos  = hist[b];
    const int S    = pos + 1;
    const float scale = 0.08838834764831845f;   // 1/sqrt(128)

    // --- RoPE on q,k; stage q/k/v rows in LDS -----------------------------
    const float* qrow = qkv + (size_t)b * NCOL + (size_t)h * D;
    const float* krow = qrow + HID;
    const float* vrow = qrow + 2 * HID;
    const float fpos = (float)pos;
#pragma unroll
    for (int t = 0; t < 2; ++t) {
        int i = lane + t * 32;                       // 0..63 (pair index)
        float freq = powf(ROPE_BASE, -(2.0f * (float)i) / (float)D);
        float ang = fpos * freq;
        float cc = cosf(ang), sn = sinf(ang);
        float q1 = qrow[i], q2 = qrow[i + D / 2];
        qs[i]         = q1 * cc - q2 * sn;
        qs[i + D / 2] = q2 * cc + q1 * sn;
        float k1 = krow[i], k2 = krow[i + D / 2];
        ks[i]         = k1 * cc - k2 * sn;
        ks[i + D / 2] = k2 * cc + k1 * sn;
        vs[i]         = vrow[i];
        vs[i + D / 2] = vrow[i + D / 2];
    }
    __syncthreads();

    // --- Pass A: scores -> LDS probs[], track local max -------------------
    float lmax = -INFINITY;
    const int nch = (S + 31) >> 5;
    for (int cch = 0; cch < nch; ++cch) {
        int s = (cch << 5) + lane;
        float sc = -INFINITY;
        if (s < S) {
            float acc = 0.0f;
            if (s == pos) {
#pragma unroll
                for (int d = 0; d < D; d += 4) {
                    float4 k4 = *(const float4*)&ks[d];
                    float4 q4 = *(const float4*)&qs[d];
                    acc += q4.x * k4.x + q4.y * k4.y + q4.z * k4.z + q4.w * k4.w;
                }
            } else {
                int blk = boff[b * NBLK + (s >> 6)];
                const float* kp = k_cache +
                    ((size_t)blk * BS + (size_t)(s & (BS - 1))) * (H * D) +
                    (size_t)h * D;
                for (int d = 0; d < D; d += 4) {
                    float4 k4 = *(const float4*)(kp + d);
                    float4 q4 = *(const float4*)&qs[d];   // LDS broadcast
                    acc += q4.x * k4.x + q4.y * k4.y + q4.z * k4.z + q4.w * k4.w;
                }
            }
            sc = acc * scale;
            probs[s] = sc;
        }
        lmax = fmaxf(lmax, sc);
    }
#pragma unroll
    for (int o = 16; o > 0; o >>= 1)
        lmax = fmaxf(lmax, __shfl_xor(lmax, o, 32));
    __syncthreads();

    // --- Pass B: exponentiate in place, local sum -------------------------
    float lsum = 0.0f;
    for (int s = lane; s < S; s += 32) {
        float p = expf(probs[s] - lmax);
        probs[s] = p;
        lsum += p;
    }
#pragma unroll
    for (int o = 16; o > 0; o >>= 1)
        lsum += __shfl_xor(lsum, o, 32);
    const float inv = 1.0f / lsum;
    __syncthreads();

    // --- Pass C: attn = probs @ V, lane owns d-chunk [4*lane, 4*lane+3] ---
    const int d0 = lane * 4;
    float4 acc = {0.f, 0.f, 0.f, 0.f};
    for (int s = 0; s < S; ++s) {
        float p = probs[s];                       // LDS broadcast read
        float4 v4;
        if (s == pos) {
            v4 = *(const float4*)&vs[d0];
        } else {
            int blk = boff[b * NBLK + (s >> 6)];
            v4 = *(const float4*)(v_cache +
                ((size_t)blk * BS + (size_t)(s & (BS - 1))) * (H * D) +
                (size_t)h * D + d0);
        }
        acc.x += p * v4.x; acc.y += p * v4.y;
        acc.z += p * v4.z; acc.w += p * v4.w;
    }
    float* op = attn_out + (size_t)b * HID + (size_t)h * D + d0;
    op[0] = acc.x * inv; op[1] = acc.y * inv;
    op[2] = acc.z * inv; op[3] = acc.w * inv;
}

// ---------------------------------------------------------------------------
// Kernel 3: out(32x4096) = attn(32x4096) @ O^T  (O is 4096x4096 row-major).
// Y[m,n] = sum_k A[m,k] * O[n,k]. WMMA f32 16x16x4, one wave per tile.
// Per-lane B reads are contiguous along k (cachelines fully consumed),
// 512 independent waves keep many HBM streams in flight.
// 2 x 256 = 512 tiles -> 64 blocks x 8 waves.
// ---------------------------------------------------------------------------
__global__ void oproj_gemm_wmma(const float* __restrict__ A,
                                const float* __restrict__ O,
                                float* __restrict__ Y)
{
    const int wave  = (blockIdx.x * blockDim.x + threadIdx.x) >> 5;
    const int lane  = threadIdx.x & 31;
    const int mtile = wave & 1;         // 0..1
    const int ntile = wave >> 1;        // 0..255
    const int mn    = lane & 15;
    const int kk    = (lane >> 4) << 1;

    const float* __restrict__ arow = A + (size_t)(mtile * 16 + mn) * HID;
    const float* __restrict__ brow = O + (size_t)(ntile * 16 + mn) * HID;

    v8f c = {};
    for (int k0 = 0; k0 < HID; k0 += 4) {
        int kpf = k0 + 64 < HID ? k0 + 64 : HID - 4;
        __builtin_prefetch(brow + kpf, 0, 1);

        v2f a = *(const v2f*)(arow + k0 + kk);
        v2f b = *(const v2f*)(brow + k0 + kk);
        c = __builtin_amdgcn_wmma_f32_16x16x4_f32(false, a, false, b,
                                                  (short)0, c, false, false);
    }

    const int rbase = mtile * 16 + ((lane >> 4) << 3);
#pragma unroll
    for (int r = 0; r < 8; ++r)
        Y[(size_t)(rbase + r) * HID + (size_t)ntile * 16 + mn] = c[r];
}

// ---------------------------------------------------------------------------
extern "C" void kernel_launch(void* const* d_in, const int* in_sizes, int n_in,
                              void* d_out, int out_size, void* d_ws, size_t ws_size,
                              hipStream_t stream)
{
    const float* X    = (const float*)d_in[0];   // hidden_states (32,4096)
    const float* Wp   = (const float*)d_in[1];   // W_pack (4096,12288)
    const float* Ow   = (const float*)d_in[2];   // o_proj_weight (4096,4096)
    const float* kc   = (const float*)d_in[3];   // k_cache
    const float* vc   = (const float*)d_in[4];   // v_cache
    const int*   hist = (const int*)d_in[5];     // history_lengths (32,)
    const int*   boff = (const int*)d_in[6];     // block_offsets (32,16)
    float* out = (float*)d_out;                  // (32,4096)

    float* qkv  = (float*)d_ws;                  // 32*12288 floats
    float* attn = qkv + (size_t)B_SZ * NCOL;     // 32*4096 floats

    // 768 strip-waves = 96 blocks * 8 waves (one 16x32 strip per wave)
    qkv_gemm_wmma<<<dim3(96), dim3(256), 0, stream>>>(X, Wp, qkv);
    // one wave per (b,h)
    attn_kernel<<<dim3(B_SZ * H), dim3(32), 0, stream>>>(qkv, attn, kc, vc, hist, boff);
    // 512 tile-waves = 64 blocks * 8 waves
    oproj_gemm_wmma<<<dim3(64), dim3(256), 0, stream>>>(attn, Ow, out);
}